// ProjectedGaussianRasterizer_56959856279982
// MI455X (gfx1250) — compile-verified
//
#include <hip/hip_runtime.h>
#include <stdint.h>

// ---------------- problem constants ----------------
#define H_      256
#define W_      256
#define G_      8192
#define B_      2
#define FRONT_K 8
#define ALPHA_T (1.0f/255.0f)
#define EPS_T   1e-8f

#define TILE    16
#define TPB     256                  // threads per raster block (8 wave32)
#define CHUNK_G 256                  // gaussians staged per chunk
#define NCHUNK  (G_/CHUNK_G)         // 32
#define GF      16                   // floats per packed gaussian (4x float4, 64B)
#define CHUNK_F4 (CHUNK_G*GF/4)      // 1024 float4 per chunk
#define F4_PER_THREAD (CHUNK_F4/TPB) // 4

#define SORT_T  1024
#define DENSE_MIN 16                 // survivors threshold for WMMA batch path

// CDNA5 async global->LDS staging (gfx1250) — verified to assemble in round 1.
#define USE_ASYNC 1

#if defined(__has_builtin)
#if __has_builtin(__builtin_amdgcn_wmma_f32_16x16x4_f32)
#define HAVE_WMMA_F32 1
#endif
#endif

typedef __attribute__((ext_vector_type(2))) float f32x2;
typedef __attribute__((ext_vector_type(8))) float f32x8;

// ---------------- async helpers ----------------
static __device__ __forceinline__ unsigned lds_off32(const void* p) {
  // generic shared-aperture address: low 32 bits are the LDS byte address
  return (unsigned)(uintptr_t)p;
}

static __device__ __forceinline__ void async_copy_b128(unsigned ldsOff, const float4* gp) {
#if USE_ASYNC
  asm volatile("global_load_async_to_lds_b128 %0, %1, off"
               :: "v"(ldsOff), "v"(gp) : "memory");
#else
  (void)ldsOff; (void)gp;
#endif
}

static __device__ __forceinline__ void wait_async4() {
#if USE_ASYNC
#if defined(__has_builtin) && __has_builtin(__builtin_amdgcn_s_wait_asynccnt)
  __builtin_amdgcn_s_wait_asynccnt(4);
#else
  asm volatile("s_wait_asynccnt 0x4" ::: "memory");
#endif
#endif
}

static __device__ __forceinline__ void wait_async0() {
#if USE_ASYNC
#if defined(__has_builtin) && __has_builtin(__builtin_amdgcn_s_wait_asynccnt)
  __builtin_amdgcn_s_wait_asynccnt(0);
#else
  asm volatile("s_wait_asynccnt 0x0" ::: "memory");
#endif
#endif
}

static __device__ __forceinline__ void stage_chunk(float4* dst /*LDS*/,
                                                   const float4* __restrict__ src,
                                                   int t) {
#pragma unroll
  for (int i = 0; i < F4_PER_THREAD; ++i) {
#if USE_ASYNC
    async_copy_b128(lds_off32(dst + t + i*TPB), src + t + i*TPB);
#else
    dst[t + i*TPB] = src[t + i*TPB];
#endif
  }
}

// ---------------- kernel 1: stable depth sort + pack ----------------
// Bitonic sort of (depth_bits<<32 | idx) keys in LDS == stable ascending depth
// order (depths >= 0 so float bits are monotone). Gather in sorted order and
// precompute per-gaussian derived data:
//   thr  = ln(255*op)+eps      (alpha prefilter: alpha>=1/255 <=> sigma<=thr)
//   rmax = sqrt(2*thr/lmin)    (conservative cull radius, lmin = min eigenvalue)
//   c0,c1,c3                   (quadratic-expansion coeffs for the WMMA path:
//                               sigma = c0 + c1*px + (a/2)px^2 + c3*py + (c/2)py^2 + b*px*py)
__global__ __launch_bounds__(SORT_T)
void gsr_sort_pack(const float* __restrict__ means2d,  // [B,G,2]
                   const float* __restrict__ conics,   // [B,G,3]
                   const float* __restrict__ colors,   // [B,G,3]
                   const float* __restrict__ opac,     // [B,G]
                   const float* __restrict__ depths,   // [B,G]
                   float* __restrict__ packed)         // [B,G,16]
{
  __shared__ unsigned long long key[G_];   // 64 KB
  const int b = blockIdx.x;
  const int t = threadIdx.x;

  for (int i = t; i < G_; i += SORT_T) {
    unsigned db = __float_as_uint(depths[b*G_ + i]);
    key[i] = ((unsigned long long)db << 32) | (unsigned)i;
  }
  __syncthreads();

  for (unsigned k = 2; k <= (unsigned)G_; k <<= 1) {
    for (unsigned j = k >> 1; j > 0; j >>= 1) {
      for (int i = t; i < G_; i += SORT_T) {
        unsigned ixj = (unsigned)i ^ j;
        if (ixj > (unsigned)i) {
          unsigned long long a = key[i], c = key[ixj];
          bool asc = ((i & k) == 0);
          if ((a > c) == asc) { key[i] = c; key[ixj] = a; }
        }
      }
      __syncthreads();
    }
  }

  for (int i = t; i < G_; i += SORT_T) {
    int src = (int)(key[i] & 0xFFFFFFFFull);
    const int gi = b*G_ + src;
    float gx = means2d[gi*2 + 0];
    float gy = means2d[gi*2 + 1];
    float ca = conics[gi*3 + 0];
    float cb = conics[gi*3 + 1];
    float cc = conics[gi*3 + 2];
    float op = opac[gi];
    float r  = colors[gi*3 + 0];
    float g  = colors[gi*3 + 1];
    float bl = colors[gi*3 + 2];
    float thr = logf(255.0f * op) + 1e-3f;
    float lmin = 0.5f*(ca + cc) - sqrtf(0.25f*(ca - cc)*(ca - cc) + cb*cb);
    float rmax = (thr > 0.0f && lmin > 0.0f) ? (sqrtf(2.0f*thr/lmin) + 0.5f)
                                             : -1.0e30f;
    float c0 = 0.5f*(ca*gx*gx + cc*gy*gy) + cb*gx*gy;
    float c1 = -(ca*gx + cb*gy);
    float c3 = -(cc*gy + cb*gx);
    float4* o = (float4*)(packed + (size_t)(b*G_ + i)*GF);
    o[0] = make_float4(gx, gy, ca, cb);
    o[1] = make_float4(cc, op, thr, rmax);
    o[2] = make_float4(r, g, bl, 0.0f);
    o[3] = make_float4(c0, c1, c3, 0.0f);
  }
}

// ---------------- kernel 2: tiled rasterization ----------------
__global__ __launch_bounds__(TPB)
void gsr_raster(const float* __restrict__ packed,  // [B,G,16]
                float* __restrict__ out)           // [B,H,W,3]
{
  __shared__ float4 buf[2][CHUNK_F4];     // 32 KB async double buffer
  __shared__ unsigned wmask[TPB/32];
  __shared__ int slist[CHUNK_G];
#if HAVE_WMMA_F32
  __shared__ float sig[TPB/32][2*272];    // per-wave 2x(16px x 16g, stride 17)
#endif

  const int t    = threadIdx.x;
  const int b    = blockIdx.x >> 8;
  const int tile = blockIdx.x & 255;
  const int tx = tile & 15, ty = tile >> 4;
  const int lx = t & 15,  ly = t >> 4;
  const int wid = t >> 5, lane = t & 31;
  const float px = (float)(tx*TILE + lx) + 0.5f;
  const float py = (float)(ty*TILE + ly) + 0.5f;
  const float cx = (float)(tx*TILE) + 8.0f;
  const float cy = (float)(ty*TILE) + 8.0f;

  const float4* src = (const float4*)packed + (size_t)b * (G_*GF/4);

  float accR = 0.0f, accG = 0.0f, accB = 0.0f, T = 1.0f;
  int count = 0;
  bool done = false;

  stage_chunk(&buf[0][0], src, t);

  for (int ch = 0; ch < NCHUNK; ++ch) {
    const int cur = ch & 1;
    if (ch + 1 < NCHUNK) {
      stage_chunk(&buf[cur ^ 1][0], src + (size_t)(ch + 1)*CHUNK_F4, t);
      wait_async4();
    } else {
      wait_async0();
    }
    __syncthreads();

    const float4* gb = &buf[cur][0];

    // --- cull: thread t tests gaussian t of this chunk against the tile ---
    float4 g0 = gb[t*4 + 0];                       // x,y,a,b
    float4 g1 = gb[t*4 + 1];                       // c,op,thr,rmax
    bool hit = (fabsf(g0.x - cx) <= g1.w + 8.0f) &&
               (fabsf(g0.y - cy) <= g1.w + 8.0f);
    unsigned m = (unsigned)__ballot(hit);          // wave32
    if (lane == 0) wmask[wid] = m;
    __syncthreads();

    // --- order-preserving compaction (depth order must survive) ---
    int ofs = 0, tot = 0;
#pragma unroll
    for (int w = 0; w < TPB/32; ++w) {
      unsigned mm = wmask[w];
      int pc = __popc(mm);
      if (w < wid) ofs += pc;
      tot += pc;
    }
    if (hit) {
      int below = (lane == 0) ? 0 : __popc(m & ((1u << lane) - 1u));
      slist[ofs + below] = t;
    }
    __syncthreads();

    // shared front-to-back composite step (identical gating to reference)
    auto composite = [&](float sigma, int gi) {
      if (sigma >= 0.0f) {
        float4 a1 = gb[gi*4 + 1];
        if (sigma <= a1.z) {
          float alpha = fminf(a1.y * __expf(-sigma), 0.999f);
          if (alpha >= ALPHA_T) {
            float4 a2 = gb[gi*4 + 2];
            if (T > EPS_T) {
              float w = T * alpha;
              accR = fmaf(w, a2.x, accR);
              accG = fmaf(w, a2.y, accG);
              accB = fmaf(w, a2.z, accB);
            }
            T *= (1.0f - alpha);
            if (++count >= FRONT_K || T <= EPS_T) done = true;
          }
        }
      }
    };

#if HAVE_WMMA_F32
    if (tot >= DENSE_MIN) {
      // ---- dense path: batched sigma via V_WMMA_F32_16X16X4_F32 ----
      // sigma[p,g] = F(p) . C(g), K=6 padded to 8 -> 2 chained WMMAs per
      // 16px x 16g tile; each wave evaluates both of its pixel rows.
      const bool lo = (lane < 16);
      const int n16 = lane & 15;
      const int half = lane >> 4;                 // 0: row 2*wid, 1: row 2*wid+1
      float* wsig = &sig[wid][0];
      const int ngrp = (tot + 15) >> 4;
      for (int g = 0; g < ngrp; ++g) {
        // B operands: coefficients of survivor n16 of this group
        int sIdx = g*16 + n16; if (sIdx > tot - 1) sIdx = tot - 1;
        int gi = slist[sIdx];
        float4 f0 = gb[gi*4 + 0];                 // x,y,a,b
        float4 f1 = gb[gi*4 + 1];                 // c,op,thr,rmax
        float4 f3 = gb[gi*4 + 3];                 // c0,c1,c3,0
        f32x2 b0, b1;
        b0.x = lo ? f3.x : 0.5f*f0.z;             // K0=c0 | K2=a/2
        b0.y = lo ? f3.y : f3.z;                  // K1=c1 | K3=c3
        b1.x = lo ? 0.5f*f1.x : 0.0f;             // K4=c/2
        b1.y = lo ? f0.w     : 0.0f;              // K5=b
#pragma unroll
        for (int rt = 0; rt < 2; ++rt) {
          float pyr = (float)(ty*TILE + 2*wid + rt) + 0.5f;
          float pxm = (float)(tx*TILE + n16) + 0.5f;   // pixel m = lane&15
          f32x2 a0, a1v;
          a0.x  = lo ? 1.0f       : pxm*pxm;      // K0 | K2
          a0.y  = lo ? pxm        : pyr;          // K1 | K3
          a1v.x = lo ? pyr*pyr    : 0.0f;         // K4
          a1v.y = lo ? pxm*pyr    : 0.0f;         // K5
          f32x8 d = {};
          d = __builtin_amdgcn_wmma_f32_16x16x4_f32(false, a0,  false, b0,
                                                    (short)0, d, false, false);
          d = __builtin_amdgcn_wmma_f32_16x16x4_f32(false, a1v, false, b1,
                                                    (short)0, d, false, false);
          // transpose D (M=pixel, N=gaussian) into per-wave LDS pad
#pragma unroll
          for (int r = 0; r < 8; ++r) {
            int mrow = r + (lo ? 0 : 8);
            wsig[rt*272 + mrow*17 + n16] = d[r];
          }
        }
        // ordered composite from staged sigmas (per-lane private row)
        int sEnd = tot - g*16; if (sEnd > 16) sEnd = 16;
        if (!done) {
          for (int n = 0; n < sEnd; ++n) {
            float sigma = wsig[half*272 + lx*17 + n];
            composite(sigma, slist[g*16 + n]);
            if (done) break;
          }
        }
      }
    } else
#endif
    {
      // ---- sparse path: scalar sigma per survivor ----
      if (!done) {
        for (int s = 0; s < tot; ++s) {
          int gi = slist[s];
          float4 a0 = gb[gi*4 + 0];
          float4 a1 = gb[gi*4 + 1];
          float dx = px - a0.x;
          float dy = py - a0.y;
          float sigma = 0.5f*((a0.z*dx)*dx + (a1.x*dy)*dy) + (a0.w*dx)*dy;
          composite(sigma, gi);
          if (done) break;
        }
      }
    }

    int alive = __syncthreads_count(done ? 0 : 1);
    if (alive == 0) break;
  }

  const size_t pix = ((size_t)b*H_ + (size_t)(ty*TILE + ly))*W_ + (tx*TILE + lx);
  float* o = out + pix*3;
  o[0] = accR; o[1] = accG; o[2] = accB;
}

// ---------------- host launcher ----------------
extern "C" void kernel_launch(void* const* d_in, const int* in_sizes, int n_in,
                              void* d_out, int out_size, void* d_ws, size_t ws_size,
                              hipStream_t stream) {
  const float* means2d = (const float*)d_in[0];
  const float* conics  = (const float*)d_in[1];
  const float* colors  = (const float*)d_in[2];
  const float* opac    = (const float*)d_in[3];
  const float* depths  = (const float*)d_in[4];
  float* packed = (float*)d_ws;   // B*G*16 floats = 1 MB

  gsr_sort_pack<<<B_, SORT_T, 0, stream>>>(means2d, conics, colors, opac, depths, packed);
  gsr_raster<<<B_*256, TPB, 0, stream>>>(packed, (float*)d_out);
}